// ROIPooling3D_80771154968757
// MI455X (gfx1250) — compile-verified
//
#include <hip/hip_runtime.h>

// ---------------- problem constants ----------------
constexpr int DD = 91, HHs = 109, WWs = 91;          // cropped atlas dims
constexpr int NVOX = DD * HHs * WWs;                 // 902629 voxels
constexpr int D0 = 96, H0 = 112, W0 = 96;            // feat spatial dims
constexpr int SPATIAL = D0 * H0 * W0;                // 1032192 per-channel stride
constexpr int BC = 128;                              // B*C = 2*64
constexpr int SEG = 95;                              // segments 0..94
constexpr int SEGSTRIDE = 129;                       // LDS pad: (id*129+bc)%64 = (id+bc)%64
constexpr int GSUM_ELEMS = SEG * BC;                 // 12160
constexpr int WS_ELEMS = GSUM_ELEMS + SEG;           // + counts
constexpr int NBLK = 768;                            // stage-1 grid

typedef float v2f __attribute__((ext_vector_type(2)));
typedef float v8f __attribute__((ext_vector_type(8)));

// ---------------- stage 0: zero the global accumulator ----------------
__global__ void roipool_zero(float* __restrict__ p, int n) {
    int i = blockIdx.x * blockDim.x + threadIdx.x;
    if (i < n) p[i] = 0.0f;
}

// ---------------- stage 1: bandwidth-bound segmented accumulation ----------------
__global__ __launch_bounds__(256) void roipool_accum(
        const float* __restrict__ feat, const int* __restrict__ atlas,
        float* __restrict__ gsum, float* __restrict__ gcnt) {
    __shared__ float acc[SEG * SEGSTRIDE];   // 12255 floats = 49020 B
    __shared__ float cnt[SEG];

    const int tid = threadIdx.x;
    for (int i = tid; i < SEG * SEGSTRIDE; i += 256) acc[i] = 0.0f;
    if (tid < SEG) cnt[tid] = 0.0f;
    __syncthreads();

    const int stride = gridDim.x * 256;
    for (int n = blockIdx.x * 256 + tid; n < NVOX; n += stride) {
        const int id = atlas[n];                     // 0..94
        const int w  = n % WWs;
        const int t  = n / WWs;
        const int h  = t % HHs;
        const int d  = t / HHs;
        const int spat = (d * H0 + h) * W0 + w;      // offset within one channel

        atomicAdd(&cnt[id], 1.0f);                   // ds_add_f32
        const float* fp = feat + spat;
        float* ap = acc + id * SEGSTRIDE;
        #pragma unroll 8
        for (int bc = 0; bc < BC; ++bc) {
            // 32 lanes = 32 consecutive w -> coalesced 128B line; stream-once -> NT
            float v = __builtin_nontemporal_load(fp + (size_t)bc * SPATIAL);
            atomicAdd(ap + bc, v);                   // ds_add_f32, conflict-free banks
        }
    }
    __syncthreads();

    // flush block-private accumulator to global (L2-resident, tiny)
    for (int i = tid; i < GSUM_ELEMS; i += 256) {
        int s = i >> 7, bc = i & (BC - 1);
        atomicAdd(&gsum[i], acc[s * SEGSTRIDE + bc]);
    }
    for (int i = tid; i < SEG; i += 256) atomicAdd(&gcnt[i], cnt[i]);
}

// ---------------- stage 2: normalize via WMMA (out = Diag(1/cnt) x sums) ----------------
// One wave handles a 16(seg) x 16(bc) tile. D = sum_{k0=0,4,8,12} A_k0 x B_k0 + C,
// with A_k0 a K-slice of the 16x16 diagonal of inverse counts, using
// V_WMMA_F32_16X16X4_F32. Exactly D[m][j] = inv[m] * sums[s0+m][bc0+j].
__global__ __launch_bounds__(256) void roipool_finalize(
        const float* __restrict__ gsum, const float* __restrict__ gcnt,
        float* __restrict__ out) {
    const int lane = threadIdx.x & 31;
    const int wv   = threadIdx.x >> 5;   // 0..7  -> bc tile
    const int st   = blockIdx.x;         // 0..5  -> segment tile
    const int s0   = st * 16;
    const int bc0  = wv * 16;
    const int m    = lane & 15;          // row (A/D) / column (B/D)
    const int hb   = lane >> 4;          // half selects K={0,1} vs K={2,3}

    const int sm = s0 + m;
    const float cv  = (sm < SEG) ? gcnt[sm] : 1.0f;
    const float inv = 1.0f / fmaxf(cv, 1e-6f);

    v8f d = {};
    #pragma unroll
    for (int k0 = 0; k0 < 16; k0 += 4) {
        const int kx = k0 + hb * 2;      // global K index of a.x / b.x
        v2f a, b;
        // A layout (16x4 f32): lane m, VGPR0 <-> K = hb?2:0 (+k0), VGPR1 <-> K+1
        a.x = (m == kx)     ? inv : 0.0f;
        a.y = (m == kx + 1) ? inv : 0.0f;
        // B layout (4x16 f32): row K striped across lanes (column = lane%16)
        const int sxa = s0 + kx;
        const int sxb = sxa + 1;
        b.x = (sxa < SEG) ? gsum[sxa * BC + bc0 + m] : 0.0f;
        b.y = (sxb < SEG) ? gsum[sxb * BC + bc0 + m] : 0.0f;
        d = __builtin_amdgcn_wmma_f32_16x16x4_f32(
                /*neg_a=*/false, a, /*neg_b=*/false, b,
                /*c_mod=*/(short)0, d, /*reuse_a=*/false, /*reuse_b=*/false);
    }

    // D layout: VGPR v -> row v (lanes 0-15) / row v+8 (lanes 16-31), col = lane%16
    #pragma unroll
    for (int v = 0; v < 8; ++v) {
        const int srow = s0 + v + hb * 8;
        if (srow >= 1 && srow <= 94) {
            const int bc = bc0 + m;
            const int b_ = bc >> 6;      // batch
            const int c_ = bc & 63;      // channel
            out[((size_t)b_ * 94 + (srow - 1)) * 64 + c_] = d[v];
        }
    }
}

// ---------------- launcher ----------------
extern "C" void kernel_launch(void* const* d_in, const int* in_sizes, int n_in,
                              void* d_out, int out_size, void* d_ws, size_t ws_size,
                              hipStream_t stream) {
    (void)in_sizes; (void)n_in; (void)out_size; (void)ws_size;
    const float* feat  = (const float*)d_in[0];
    const int*   atlas = (const int*)d_in[1];
    float* out  = (float*)d_out;
    float* gsum = (float*)d_ws;              // 12160 floats
    float* gcnt = gsum + GSUM_ELEMS;         // 95 floats

    roipool_zero<<<(WS_ELEMS + 255) / 256, 256, 0, stream>>>(gsum, WS_ELEMS);
    roipool_accum<<<NBLK, 256, 0, stream>>>(feat, atlas, gsum, gcnt);
    roipool_finalize<<<6, 256, 0, stream>>>(gsum, gcnt, out);
}